// Mamba2Layer_30039001268739
// MI455X (gfx1250) — compile-verified
//
#include <hip/hip_runtime.h>

typedef float v2f __attribute__((ext_vector_type(2)));
typedef float v8f __attribute__((ext_vector_type(8)));

constexpr int D_MODEL  = 1024;
constexpr int D_STATE  = 128;
constexpr int D_INNER  = 2048;
constexpr int PROJ_DIM = 2 * D_INNER + 2 * D_STATE + 1; // 4353
constexpr int NB       = 4;
constexpr int TT       = 4096;
constexpr int M_ROWS   = NB * TT; // 16384
constexpr float CLIPV  = 8.0f;
constexpr float EPSV   = 1e-5f;

__device__ __forceinline__ float csilu(float v) {
  float s = v / (1.0f + __expf(-v));
  return fminf(s, CLIPV);
}
__device__ __forceinline__ float softplusf(float v) {
  return v > 20.0f ? v : log1pf(__expf(v));
}

// ---------------------------------------------------------------------------
// C[M x N] = A[M x K] * W[N x K]^T  using V_WMMA_F32_16X16X4_F32.
// One wave computes a 16(M) x 64(N) strip: 4 accumulators, A fragment reused.
// Double-buffered k-blocks of 8 (2 wmma steps): 10 staged fragments per
// phase (40 VGPRs total staging) -> no spills, loads batch ahead of the
// 8-WMMA compute phase with non-zero s_wait_loadcnt thresholds.
//
// fp32 WMMA fragment layout (ISA 7.12.2):
//   A 16x4 : lane L -> row M = L&15, holds K = {2*(L>>4), 2*(L>>4)+1}
//   B 4x16 : lane L -> col N = L&15, holds K = {2*(L>>4), 2*(L>>4)+1}
//   C/D    : lane L -> col N = L&15, VGPR v -> row M = v + 8*(L>>4)
// ---------------------------------------------------------------------------
__global__ void k_wmma_gemm(const float* __restrict__ A, int K,
                            const float* __restrict__ W,
                            float* __restrict__ C, int N, int tilesN)
{
  int wave  = (int)((blockIdx.x * blockDim.x + threadIdx.x) >> 5);
  int lane  = threadIdx.x & 31;
  int tileM = wave / tilesN;
  int tileN = wave % tilesN;

  int mrow = tileM * 16 + (lane & 15);
  int kb   = (lane >> 4) << 1;                 // k sub-offset {0,2}
  const float* arow = A + (size_t)mrow * K + kb;

  const float* wrow[4];
  int ncol[4];
#pragma unroll
  for (int j = 0; j < 4; ++j) {
    int n   = tileN * 64 + j * 16 + (lane & 15);
    ncol[j] = n;
    int nc  = (n < N) ? n : (N - 1);           // clamp loads; mask stores
    wrow[j] = W + (size_t)nc * K + kb;
  }

  v8f acc[4] = {};
  v2f afrag[2][2];
  v2f bfrag[2][2][4];

  // Load one k-block (8 k values = 2 wmma steps) into staging registers.
  auto loadblk = [&](int k0, int p) {
#pragma unroll
    for (int u = 0; u < 2; ++u) {
      afrag[p][u] = *(const v2f*)(arow + k0 + u * 4);
#pragma unroll
      for (int j = 0; j < 4; ++j)
        bfrag[p][u][j] = *(const v2f*)(wrow[j] + k0 + u * 4);
    }
  };
  auto compute = [&](int p) {
#pragma unroll
    for (int u = 0; u < 2; ++u) {
#pragma unroll
      for (int j = 0; j < 4; ++j) {
        acc[j] = __builtin_amdgcn_wmma_f32_16x16x4_f32(
            /*neg_a=*/false, afrag[p][u], /*neg_b=*/false, bfrag[p][u][j],
            /*c_mod=*/(short)0, acc[j], /*reuse_a=*/false, /*reuse_b=*/false);
      }
    }
  };

  // K is a multiple of 16 for all uses (1024, 2048).
  loadblk(0, 0);
  for (int k0 = 0; k0 < K; k0 += 16) {
    loadblk(k0 + 8, 1);
    compute(0);
    if (k0 + 16 < K) loadblk(k0 + 16, 0);
    compute(1);
  }

  int mbase = tileM * 16 + ((lane >> 4) << 3);
#pragma unroll
  for (int j = 0; j < 4; ++j) {
    if (ncol[j] < N) {
#pragma unroll
      for (int v = 0; v < 8; ++v)
        C[(size_t)(mbase + v) * N + ncol[j]] = acc[j][v];
    }
  }
}

// ---------------------------------------------------------------------------
// Depthwise causal conv (width 4) over time with csilu on input and output.
// One thread per (m, d).
// ---------------------------------------------------------------------------
__global__ void k_conv(const float* __restrict__ zxbcdt,
                       const float* __restrict__ conv_w,
                       float* __restrict__ xconv)
{
  long idx = (long)blockIdx.x * blockDim.x + threadIdx.x;
  int d = (int)(idx & (D_INNER - 1));
  int m = (int)(idx >> 11);
  if (m >= M_ROWS) return;
  int t = m & (TT - 1);

  float w[4];
#pragma unroll
  for (int i = 0; i < 4; ++i) w[i] = conv_w[d * 4 + i];

  float acc = 0.f;
#pragma unroll
  for (int i = 0; i < 4; ++i) {
    int tt = t - 3 + i;
    if (tt >= 0) {
      float xv = zxbcdt[(size_t)(m - 3 + i) * PROJ_DIM + D_INNER + d];
      acc = fmaf(w[i], csilu(xv), acc);
    }
  }
  xconv[idx] = csilu(acc);
}

// ---------------------------------------------------------------------------
// Pack per-(t,s) scan coefficients: (A_bar, B_bar, C, 0) as float4 so the
// scan inner loop is one b128 broadcast load per state.
// ---------------------------------------------------------------------------
__global__ void k_coef(const float* __restrict__ zxbcdt,
                       const float* __restrict__ A_log,
                       float4* __restrict__ coef)
{
  int idx = blockIdx.x * blockDim.x + threadIdx.x; // m * 128 + s
  int s = idx & (D_STATE - 1);
  int m = idx >> 7;
  if (m >= M_ROWS) return;
  const float* row = zxbcdt + (size_t)m * PROJ_DIM;
  float dtv  = softplusf(row[PROJ_DIM - 1]);
  float a    = -__expf(A_log[s]);
  float abar = __expf(a * dtv);
  float bbar = row[2 * D_INNER + s] * dtv;
  float c    = row[2 * D_INNER + D_STATE + s];
  coef[idx] = make_float4(abar, bbar, c, 0.f);
}

// ---------------------------------------------------------------------------
// Sequential selective scan. One lane per (b, d); h[128] lives in VGPRs.
// 32 blocks x 256 threads = 8192 lanes = B * D_INNER.
// ---------------------------------------------------------------------------
__global__ void k_scan(const float4* __restrict__ coef,
                       const float* __restrict__ xconv,
                       const float* __restrict__ Dp,
                       float* __restrict__ ybuf,
                       float* __restrict__ hT)
{
  int b = blockIdx.x >> 3;
  int d = ((blockIdx.x & 7) << 8) + threadIdx.x;

  float h[D_STATE];
#pragma unroll
  for (int s = 0; s < D_STATE; ++s) h[s] = 0.f;
  float Dd = Dp[d];

  for (int t = 0; t < TT; ++t) {
    int m = b * TT + t;
    float xt = xconv[(size_t)m * D_INNER + d];
    const float4* cf = coef + (size_t)m * D_STATE;
    float y = Dd * xt;
#pragma unroll
    for (int s = 0; s < D_STATE; ++s) {
      float4 c4 = cf[s];
      h[s] = fmaf(c4.x, h[s], c4.y * xt);
      y    = fmaf(h[s], c4.z, y);
    }
    ybuf[(size_t)m * D_INNER + d] = y;
  }

  float* hrow = hT + ((size_t)b * D_INNER + d) * D_STATE;
#pragma unroll
  for (int s = 0; s < D_STATE; ++s) hrow[s] = h[s];
}

// ---------------------------------------------------------------------------
// RMS-norm over d (2048) + norm_w + csilu(z) gate, in place on ybuf.
// One block (256 threads) per row m.
// ---------------------------------------------------------------------------
__global__ void k_norm_gate(const float* __restrict__ zxbcdt,
                            float* __restrict__ ybuf,
                            const float* __restrict__ norm_w)
{
  int m = blockIdx.x;
  int tid = threadIdx.x;
  __shared__ float red[256];

  float* yrow = ybuf + (size_t)m * D_INNER;
  float ss = 0.f;
#pragma unroll
  for (int i = 0; i < 8; ++i) {
    float v = yrow[tid + i * 256];
    ss = fmaf(v, v, ss);
  }
  red[tid] = ss;
  __syncthreads();
  for (int off = 128; off > 0; off >>= 1) {
    if (tid < off) red[tid] += red[tid + off];
    __syncthreads();
  }
  float rstd = rsqrtf(red[0] * (1.0f / D_INNER) + EPSV);

  const float* zrow = zxbcdt + (size_t)m * PROJ_DIM;
#pragma unroll
  for (int i = 0; i < 8; ++i) {
    int d = tid + i * 256;
    yrow[d] = yrow[d] * rstd * norm_w[d] * csilu(zrow[d]);
  }
}

// ---------------------------------------------------------------------------
extern "C" void kernel_launch(void* const* d_in, const int* in_sizes, int n_in,
                              void* d_out, int out_size, void* d_ws, size_t ws_size,
                              hipStream_t stream)
{
  (void)in_sizes; (void)n_in; (void)out_size; (void)ws_size;
  const float* x    = (const float*)d_in[0]; // [4,4096,1024]
  const float* w1   = (const float*)d_in[1]; // [4353,1024]
  const float* cw   = (const float*)d_in[2]; // [2048,1,4]
  const float* alog = (const float*)d_in[3]; // [128]
  const float* Dp   = (const float*)d_in[4]; // [2048]
  const float* nw   = (const float*)d_in[5]; // [2048]
  const float* w2   = (const float*)d_in[6]; // [1024,2048]

  float* out = (float*)d_out;                        // [4,4096,1024]
  float* hT  = out + (size_t)NB * TT * D_MODEL;      // [4,2048,128]

  char* ws = (char*)d_ws;
  size_t off = 0;
  float*  zxbcdt = (float*)(ws + off);  off += (size_t)M_ROWS * PROJ_DIM * 4;  // 285 MB
  float*  xconv  = (float*)(ws + off);  off += (size_t)M_ROWS * D_INNER * 4;   // 134 MB
  float*  ybuf   = (float*)(ws + off);  off += (size_t)M_ROWS * D_INNER * 4;   // 134 MB
  float4* coef   = (float4*)(ws + off); off += (size_t)M_ROWS * D_STATE * 16;  // 33.5 MB

  // 1. in_proj GEMM: zxbcdt = x @ in_proj_w^T   (M=16384, K=1024, N=4353)
  {
    int tilesN = (PROJ_DIM + 63) / 64;            // 69
    int waves  = (M_ROWS / 16) * tilesN;          // 70656
    int blocks = (waves + 7) / 8;                 // 8832 (exact)
    k_wmma_gemm<<<blocks, 256, 0, stream>>>(x, D_MODEL, w1, zxbcdt, PROJ_DIM, tilesN);
  }

  // 2. depthwise causal conv + csilu
  k_conv<<<(M_ROWS * D_INNER) / 256, 256, 0, stream>>>(zxbcdt, cw, xconv);

  // 3. pack scan coefficients (A_bar, B_bar, C)
  k_coef<<<(M_ROWS * D_STATE) / 256, 256, 0, stream>>>(zxbcdt, alog, coef);

  // 4. sequential selective scan (also writes hT output)
  k_scan<<<NB * 8, 256, 0, stream>>>(coef, xconv, Dp, ybuf, hT);

  // 5. RMS-norm + gate
  k_norm_gate<<<M_ROWS, 256, 0, stream>>>(zxbcdt, ybuf, nw);

  // 6. out_proj GEMM: out = yg @ out_proj_w^T   (M=16384, K=2048, N=1024)
  {
    int tilesN = D_MODEL / 64;                    // 16
    int waves  = (M_ROWS / 16) * tilesN;          // 16384
    k_wmma_gemm<<<waves / 8, 256, 0, stream>>>(ybuf, D_INNER, w2, out, D_MODEL, tilesN);
  }
}